// algo_amp_36936718745675
// MI455X (gfx1250) — compile-verified
//
#include <hip/hip_runtime.h>
#include <hip/hip_bf16.h>
#include <math.h>
#include <stdint.h>

#define B_TOT    4096
#define NRX      128
#define NTX      64
#define NUM_ITER 64
#define NBITS    2

#define HT_STRIDE 72   // 64 cols + 8 pad -> half-wave row groups hit disjoint LDS banks
#define HH_STRIDE 65   // conflict-free column reads in the iteration matvec

typedef __attribute__((ext_vector_type(2))) float v2f;
typedef __attribute__((ext_vector_type(8))) float v8f;
typedef __attribute__((ext_vector_type(4))) unsigned int v4u;
typedef __attribute__((ext_vector_type(8))) int v8i;
typedef __attribute__((ext_vector_type(4))) int v4i;

struct Post { float w0, w1, w2, w3, s; };

// PAM-4 points m = {-3,-1,1,3}/sqrt(5)
#define C5 0.4472135954999579f
#define M0 (-3.0f * C5)
#define M1 (-1.0f * C5)
#define M2 ( 1.0f * C5)
#define M3 ( 3.0f * C5)

// y_scale = sqrt(2/128) = 0.125 exactly; fold y_scale^2 = 2^-6 (exact power of two)
#define HSCALE2 0.015625f

#if defined(__has_builtin)
#if __has_builtin(__builtin_amdgcn_tensor_load_to_lds)
#define HAVE_TDM 1
#endif
#endif
#ifndef HAVE_TDM
#define HAVE_TDM 0
#endif

__device__ __forceinline__ Post posterior_step(float z, float ninv2v) {
    float a0 = ninv2v * (z - M0) * (z - M0);
    float a1 = ninv2v * (z - M1) * (z - M1);
    float a2 = ninv2v * (z - M2) * (z - M2);
    float a3 = ninv2v * (z - M3) * (z - M3);
    float mx = fmaxf(fmaxf(a0, a1), fmaxf(a2, a3));
    float w0 = expf(a0 - mx), w1 = expf(a1 - mx);
    float w2 = expf(a2 - mx), w3 = expf(a3 - mx);
    float inv = 1.0f / (w0 + w1 + w2 + w3);
    w0 *= inv; w1 *= inv; w2 *= inv; w3 *= inv;
    Post p;
    p.w0 = w0; p.w1 = w1; p.w2 = w2; p.w3 = w3;
    p.s = w0 * M0 + w1 * M1 + w2 * M2 + w3 * M3;
    return p;
}

__launch_bounds__(256)
__global__ void amp_pam4_kernel(const float* __restrict__ y,
                                const float* __restrict__ h,
                                const float* __restrict__ sigma,
                                const float* __restrict__ gamma,
                                const float* __restrict__ delta,
                                float* __restrict__ out)
{
    __shared__ float ldsHt[NRX * HT_STRIDE];   // 36864 B, RAW h[b], row-major [n][k], 72-stride
    __shared__ float ldsHH[NTX * HH_STRIDE];   // 16640 B, HH = Ht^T Ht (scaled)
    __shared__ float ldsYt[NRX];               // raw y[b]
    __shared__ float ldsS[NTX];
    __shared__ float ldsG[NTX];
    __shared__ float ldsRed[4 * NTX];
    __shared__ float ldsGam[NUM_ITER + 1];
    __shared__ float ldsDel[NUM_ITER];
    __shared__ float ldsGsum;

    const int t = threadIdx.x;
    const int b = blockIdx.x;

    const float beta     = (float)NTX / (float)NRX;                   // 0.5
    const float mean_am2 = 0.0625f * (M0*M0 + M1*M1 + M2*M2 + M3*M3); // 0.25

    const float sig = sigma[b];
    const float N0  = sig * sig;

#if HAVE_TDM
    // ---- stage raw h[b] (128 x 64 fp32 tile) into LDS via the Tensor Data Mover ----
    // D# per CDNA5 ISA: 2D tensor, data_size=4B, tile 64(X) x 128(Y),
    // hardware row padding: pad_interval=5 (every 64 dwords) + pad_amount=7 (8 dwords)
    // reproduces the 72-float LDS row stride.  Wave 0 issues the single DMA.
    if (t < 32) {
        uint64_t ga = (uint64_t)(uintptr_t)(h + (size_t)b * NRX * NTX);
        uint32_t la = (uint32_t)(uintptr_t)&ldsHt[0];   // low 32 bits = LDS byte offset
        v4u g0 = { 1u,                                           // count=1, user descriptor
                   la,                                           // lds_addr [63:32]
                   (uint32_t)ga,                                 // global_addr [95:64]
                   ((uint32_t)(ga >> 32) & 0x01FFFFFFu)          // global_addr [120:96]
                   | (2u << 30) };                               // type=2 ("image")
        v8i g1 = { (int)((2u << 16)        // data_size = 4 bytes
                 | (1u << 20)              // pad_enable
                 | (5u << 22)              // pad_interval: 2^(5+1)=64 dwords
                 | (7u << 25)),            // pad_amount: 7+1 = 8 dwords
                   (int)(64u  << 16),      // tensor_dim0 = 64   (bits 79:48, low half)
                   (int)(128u << 16),      // tensor_dim1 = 128  (bits 111:80, low half)
                   (int)(64u  << 16),      // tile_dim0 = 64     (bits 127:112)
                   (int)128u,              // tile_dim1 = 128    (bits 143:128)
                   (int)64u,               // tensor_dim0_stride = 64 (bits 207:160, low)
                   0, 0 };
        v4i gz4 = { 0, 0, 0, 0 };          // groups 2/3 unused (2D tensor)
        v8i gz8 = { 0, 0, 0, 0, 0, 0, 0, 0 };
        __builtin_amdgcn_tensor_load_to_lds(g0, g1, gz4, gz4, gz8, 0);
    }
#else
    // fallback: manual coalesced copy of raw h[b]
    {
        const float4* h4 = (const float4*)(h + (size_t)b * NRX * NTX);
        #pragma unroll
        for (int i = 0; i < (NRX * NTX / 4) / 256; ++i) {
            int idx4 = t + i * 256;
            int n = idx4 >> 4;
            int k = (idx4 & 15) << 2;
            float4 v = h4[idx4];
            float* dst = &ldsHt[n * HT_STRIDE + k];
            dst[0] = v.x; dst[1] = v.y; dst[2] = v.z; dst[3] = v.w;
        }
    }
#endif

    if (t < NUM_ITER + 1) ldsGam[t] = gamma[t];
    if (t < NUM_ITER)     ldsDel[t] = delta[t];
    if (t < NRX)          ldsYt[t]  = y[(size_t)b * NRX + t];

#if HAVE_TDM
    if (t < 32) __builtin_amdgcn_s_wait_tensorcnt(0);
#endif
    __syncthreads();

    // ---- HH = y_scale^2 * (h^T h) via V_WMMA_F32_16X16X4_F32 ----
    // 16 output tiles (4x4 grid of 16x16); 8 waves, 2 tiles each (shared A operand).
    {
        const int wave = t >> 5;
        const int lane = t & 31;
        const int lo   = lane & 15;
        const int hi   = lane >> 4;
        const int ti   = wave >> 1;          // row tile 0..3
        const int tj   = (wave & 1) << 1;    // col tile base 0 or 2
        const int colA  = ti * 16 + lo;
        const int colB0 = tj * 16 + lo;
        const int colB1 = (tj + 1) * 16 + lo;

        v8f c0 = {};
        v8f c1 = {};
        for (int k0 = 0; k0 < NRX; k0 += 4) {
            // A(16x4): lane holds M=lo, K = 2*hi + vgpr ; B(4x16): N=lo, K = vgpr + 2*hi
            const float* row0 = &ldsHt[(k0 + 2 * hi) * HT_STRIDE];
            const float* row1 = row0 + HT_STRIDE;
            v2f a, b0, b1;
            a.x  = row0[colA];   a.y  = row1[colA];
            b0.x = row0[colB0];  b0.y = row1[colB0];
            b1.x = row0[colB1];  b1.y = row1[colB1];
            c0 = __builtin_amdgcn_wmma_f32_16x16x4_f32(false, a, false, b0, (short)0, c0, false, false);
            c1 = __builtin_amdgcn_wmma_f32_16x16x4_f32(false, a, false, b1, (short)0, c1, false, false);
        }
        #pragma unroll
        for (int r = 0; r < 8; ++r) {
            int row = ti * 16 + r + 8 * hi;   // C/D layout: VGPR r = rows r and r+8
            ldsHH[row * HH_STRIDE + tj * 16 + lo]       = HSCALE2 * c0[r];
            ldsHH[row * HH_STRIDE + (tj + 1) * 16 + lo] = HSCALE2 * c1[r];
        }
    }
    __syncthreads();

    // ---- yH and initial AMP state (fold y_scale^2 exactly) ----
    float yH_t = 0.0f, s_t = 0.0f, rH_t = 0.0f;
    if (t < NTX) {
        float acc = 0.0f;
        for (int n = 0; n < NRX; ++n)
            acc = fmaf(ldsYt[n], ldsHt[n * HT_STRIDE + t], acc);
        yH_t = HSCALE2 * acc;
        s_t  = 0.0f;    // alpha @ m == 0 exactly (symmetric PAM, uniform priors)
        rH_t = yH_t;    // yH - HH @ 0
    }
    float tau = beta * mean_am2 / N0;

    const int mk = t & 63;   // matvec output row
    const int mq = t >> 6;   // matvec quarter of the K range

    for (int it = 0; it < NUM_ITER; ++it) {
        const float g = ldsGam[it];
        const float d = ldsDel[it];
        const float tau_g  = tau * g;
        const float ninv2v = -0.5f / (N0 * (1.0f + tau_g));

        if (t < NTX) {
            Post p = posterior_step(s_t + rH_t, ninv2v);
            s_t = p.s;
            float G = p.w0 * (M0 - p.s) * (M0 - p.s)
                    + p.w1 * (M1 - p.s) * (M1 - p.s)
                    + p.w2 * (M2 - p.s) * (M2 - p.s)
                    + p.w3 * (M3 - p.s) * (M3 - p.s);
            ldsS[t] = p.s;
            ldsG[t] = G;
        }
        __syncthreads();

        // wave 0: reduce sum(G) over 64 symbols
        if (t < 32) {
            float v = ldsG[t] + ldsG[t + 32];
            for (int off = 16; off > 0; off >>= 1) v += __shfl_xor(v, off);
            if (t == 0) ldsGsum = v;
        }
        // all 256 threads: split matvec HH @ s_new  (row mk, K-quarter mq)
        {
            const float* hrow = &ldsHH[mk * HH_STRIDE + mq * 16];
            const float* sp   = &ldsS[mq * 16];
            float acc = 0.0f;
            #pragma unroll
            for (int l = 0; l < 16; ++l) acc = fmaf(hrow[l], sp[l], acc);
            ldsRed[mq * NTX + mk] = acc;
        }
        __syncthreads();

        const float tau_old = tau_g * d;
        const float tau_new = (beta / N0) * (ldsGsum * (1.0f / NTX));
        if (t < NTX) {
            float mv = ldsRed[t] + ldsRed[NTX + t] + ldsRed[2 * NTX + t] + ldsRed[3 * NTX + t];
            rH_t = yH_t - d * mv + (tau_new / (tau_old + 1.0f)) * rH_t;
        }
        tau = tau_new;
    }

    // ---- final posterior (gamma[NUM_ITER]) + LLRs ----
    {
        const float tau_g  = tau * ldsGam[NUM_ITER];
        const float ninv2v = -0.5f / (N0 * (1.0f + tau_g));
        if (t < NTX) {
            Post p = posterior_step(s_t + rH_t, ninv2v);
            // bit c of symbol index j (LSB-first): bit0 -> j in {1,3}; bit1 -> j in {2,3}
            float p_b0 = p.w1 + p.w3;
            float p_b1 = p.w2 + p.w3;
            float llr0 = logf(p_b0 / (1.0f - p_b0 + 1e-20f));
            float llr1 = logf(p_b1 / (1.0f - p_b1 + 1e-20f));
            llr0 = fminf(fmaxf(llr0, -1e9f), 1e9f);
            llr1 = fminf(fmaxf(llr1, -1e9f), 1e9f);
            size_t base = (size_t)b * NTX + t;
            out[base * NBITS + 0] = llr0;
            out[base * NBITS + 1] = llr1;
            out[(size_t)B_TOT * NTX * NBITS + base] = p.s;
        }
    }
}

extern "C" void kernel_launch(void* const* d_in, const int* in_sizes, int n_in,
                              void* d_out, int out_size, void* d_ws, size_t ws_size,
                              hipStream_t stream) {
    (void)in_sizes; (void)n_in; (void)out_size; (void)d_ws; (void)ws_size;
    const float* y     = (const float*)d_in[0];
    const float* h     = (const float*)d_in[1];
    const float* sigma = (const float*)d_in[2];
    const float* gamma = (const float*)d_in[3];
    const float* delta = (const float*)d_in[4];
    float* out = (float*)d_out;
    amp_pam4_kernel<<<B_TOT, 256, 0, stream>>>(y, h, sigma, gamma, delta, out);
}